// GINNet_21165598834941
// MI455X (gfx1250) — compile-verified
//
#include <hip/hip_runtime.h>

typedef __attribute__((ext_vector_type(2))) float v2f;
typedef __attribute__((ext_vector_type(8))) float v8f;

#define HDIM 64

// ---------------------------------------------------------------------------
// Edge aggregation: agg[dst[e], :] += feat[src[e], :]   (segment_sum of x[src])
// CHUNKS = F/4 is a compile-time power of two so e/c indexing is shift/mask.
// Consecutive lanes take consecutive float4 chunks of the same edge ->
// coalesced gather and coalesced global_atomic_add_f32.
// ---------------------------------------------------------------------------
template <int CHUNKS>
__global__ __launch_bounds__(256) void gin_scatter_kernel(
    const float* __restrict__ feat, const int* __restrict__ src,
    const int* __restrict__ dst, float* __restrict__ agg, int nedges)
{
  constexpr int F = CHUNKS * 4;
  const long long total  = (long long)nedges * CHUNKS;
  const long long stride = (long long)gridDim.x * blockDim.x;
  for (long long idx = (long long)blockIdx.x * blockDim.x + threadIdx.x;
       idx < total; idx += stride) {
    const int e = (int)(idx / CHUNKS);            // shift (CHUNKS = 2^k)
    const int c = ((int)idx & (CHUNKS - 1)) << 2; // mask
    const int s = src[e];
    const int d = dst[e];
    const float4 v = *(const float4*)(feat + (size_t)s * F + c);
    float* p = agg + (size_t)d * F + c;
    atomicAdd(p + 0, v.x);
    atomicAdd(p + 1, v.y);
    atomicAdd(p + 2, v.z);
    atomicAdd(p + 3, v.w);
  }
}

// ---------------------------------------------------------------------------
// Fused GIN MLP for one layer:
//   out = relu( relu( (agg + xin) @ w1 + b1 ) @ w2 + b2 )
// 8 waves / block, each wave owns a 16-row node tile and computes the full
// [16, 64] output with V_WMMA_F32_16X16X4_F32 (exact f32 path).
//
// w2 + biases are staged block-wide into LDS with GLOBAL_LOAD_ASYNC_TO_LDS_B128
// (ASYNCcnt) before any wave diverges; the hidden tile round-trips through
// per-wave LDS to convert the C/D layout into the A layout between GEMMs.
//
// Fragment layouts (wave32, per CDNA5 ISA 7.12.2):
//   A (16x4):  row M = lane%16; lanes 0-15 hold K={0,1}, lanes 16-31 K={2,3}
//   B (4x16):  col N = lane%16; lanes 0-15 hold K={0,1}, lanes 16-31 K={2,3}
//   C/D (16x16): N = lane%16; vgpr r holds M = r + 8*(lane/16)
// ---------------------------------------------------------------------------
__global__ __launch_bounds__(256) void gin_mlp_kernel(
    const float* __restrict__ agg, const float* __restrict__ xin,
    const float* __restrict__ w1, const float* __restrict__ b1,
    const float* __restrict__ w2, const float* __restrict__ b2,
    float* __restrict__ out, int nrows, int K)
{
  __shared__ float h_lds[8][16 * HDIM];       // 32 KB: per-wave hidden tiles
  __shared__ float w2_lds[HDIM * HDIM];       // 16 KB: second-layer weights
  __shared__ float b1_lds[HDIM];
  __shared__ float b2_lds[HDIM];

  // ---- async stage of w2 (+ biases) into LDS; all waves participate ----
  {
    const float* gsrc = w2 + threadIdx.x * 4;
#pragma unroll
    for (int i = 0; i < (HDIM * HDIM / 4) / 256; ++i) {
      // flat->LDS aperture: low 32 bits of the flat address are the LDS byte
      // address (ISA 10.2); feed it as the per-lane VDST of the async copy.
      unsigned int ldst =
          (unsigned int)(unsigned long long)&w2_lds[(i * 256 + threadIdx.x) * 4];
      asm volatile("global_load_async_to_lds_b128 %0, %1, off"
                   :: "v"(ldst), "v"(gsrc + i * 256 * 4) : "memory");
    }
    if (threadIdx.x < HDIM / 4) {
      unsigned int l1 =
          (unsigned int)(unsigned long long)&b1_lds[threadIdx.x * 4];
      unsigned int l2 =
          (unsigned int)(unsigned long long)&b2_lds[threadIdx.x * 4];
      asm volatile("global_load_async_to_lds_b128 %0, %1, off"
                   :: "v"(l1), "v"(b1 + threadIdx.x * 4) : "memory");
      asm volatile("global_load_async_to_lds_b128 %0, %1, off"
                   :: "v"(l2), "v"(b2 + threadIdx.x * 4) : "memory");
    }
    asm volatile("s_wait_asynccnt 0" ::: "memory");
  }
  __syncthreads();

  const int lane = threadIdx.x & 31;
  const int wave = threadIdx.x >> 5;
  const int tile = blockIdx.x * 8 + wave;
  const int row0 = tile * 16;
  if (row0 >= nrows) return;  // wave-uniform exit; EXEC stays all-ones

  const int mrow  = lane & 15;
  const int khalf = lane >> 4;  // 0: K pair {0,1}; 1: K pair {2,3}
  int grow = row0 + mrow;
  if (grow >= nrows) grow = nrows - 1;  // clamp loads on tail tile

  const float* arow = agg + (size_t)grow * K;
  const float* xrow = xin + (size_t)grow * K;

  // ---- GEMM 1: [16,K] x [K,64] -> 4 column tiles of 16x16 ----
  v8f acc[4] = {};
  const int ksteps = K >> 2;
  for (int ks = 0; ks < ksteps; ++ks) {
    const int kb = ks * 4 + khalf * 2;
    v2f a;
    a.x = arow[kb]     + xrow[kb];
    a.y = arow[kb + 1] + xrow[kb + 1];
#pragma unroll
    for (int c = 0; c < 4; ++c) {
      const int n = c * 16 + mrow;
      v2f b;
      b.x = w1[(size_t)kb * HDIM + n];
      b.y = w1[(size_t)(kb + 1) * HDIM + n];
      acc[c] = __builtin_amdgcn_wmma_f32_16x16x4_f32(
          false, a, false, b, (short)0, acc[c], false, false);
    }
  }

  // bias + ReLU; write hidden [16,64] tile to per-wave LDS (layout reshuffle)
  float* hw = &h_lds[wave][0];
#pragma unroll
  for (int c = 0; c < 4; ++c) {
    const int n = c * 16 + mrow;
    const float bias = b1_lds[n];
#pragma unroll
    for (int r = 0; r < 8; ++r) {
      const int m = r + khalf * 8;
      const float v = acc[c][r] + bias;
      hw[m * HDIM + n] = fmaxf(v, 0.0f);
    }
  }
  // LDS ops from the same wave are in-order (DScnt); tile is wave-private.

  // ---- GEMM 2: [16,64] x [64,64], B fragments from LDS ----
  v8f acc2[4] = {};
#pragma unroll 4
  for (int ks = 0; ks < (HDIM >> 2); ++ks) {
    const int kb = ks * 4 + khalf * 2;
    const v2f a = *(const v2f*)&hw[mrow * HDIM + kb];  // 8B-aligned ds_load
#pragma unroll
    for (int c = 0; c < 4; ++c) {
      const int n = c * 16 + mrow;
      v2f b;
      b.x = w2_lds[kb * HDIM + n];
      b.y = w2_lds[(kb + 1) * HDIM + n];
      acc2[c] = __builtin_amdgcn_wmma_f32_16x16x4_f32(
          false, a, false, b, (short)0, acc2[c], false, false);
    }
  }

  // bias + ReLU + store
#pragma unroll
  for (int c = 0; c < 4; ++c) {
    const int n = c * 16 + mrow;
    const float bias = b2_lds[n];
#pragma unroll
    for (int r = 0; r < 8; ++r) {
      const int m  = r + khalf * 8;
      const int gr = row0 + m;
      if (gr < nrows) {
        const float v = acc2[c][r] + bias;
        out[(size_t)gr * HDIM + n] = fmaxf(v, 0.0f);
      }
    }
  }
}

// ---------------------------------------------------------------------------
// Global add pool: pooled[batch[i], :] += h[i, :]
// ---------------------------------------------------------------------------
__global__ __launch_bounds__(256) void gin_pool_kernel(
    const float* __restrict__ h, const int* __restrict__ batch,
    float* __restrict__ pooled, int nnodes)
{
  constexpr int CHUNKS = HDIM / 4;  // 16
  const long long total  = (long long)nnodes * CHUNKS;
  const long long stride = (long long)gridDim.x * blockDim.x;
  for (long long idx = (long long)blockIdx.x * blockDim.x + threadIdx.x;
       idx < total; idx += stride) {
    const int node = (int)(idx / CHUNKS);
    const int c    = ((int)idx & (CHUNKS - 1)) << 2;
    const int g    = batch[node];
    const float4 v = *(const float4*)(h + (size_t)node * HDIM + c);
    float* p = pooled + (size_t)g * HDIM + c;
    atomicAdd(p + 0, v.x);
    atomicAdd(p + 1, v.y);
    atomicAdd(p + 2, v.z);
    atomicAdd(p + 3, v.w);
  }
}

// ---------------------------------------------------------------------------
// Readout: out[g] = pooled[g,:] . w_out + b_out
// ---------------------------------------------------------------------------
__global__ __launch_bounds__(256) void gin_readout_kernel(
    const float* __restrict__ pooled, const float* __restrict__ w_out,
    const float* __restrict__ b_out, float* __restrict__ out, int G)
{
  const int g = blockIdx.x * blockDim.x + threadIdx.x;
  if (g >= G) return;
  float s = 0.0f;
#pragma unroll
  for (int i = 0; i < HDIM; ++i) s += pooled[(size_t)g * HDIM + i] * w_out[i];
  out[g] = s + b_out[0];
}

// ---------------------------------------------------------------------------
extern "C" void kernel_launch(void* const* d_in, const int* in_sizes, int n_in,
                              void* d_out, int out_size, void* d_ws, size_t ws_size,
                              hipStream_t stream)
{
  (void)n_in; (void)ws_size;

  const float* x      = (const float*)d_in[0];
  const int*   eidx   = (const int*)d_in[1];
  // d_in[2] = edge_attr (unused by the reference network)
  const int*   batch  = (const int*)d_in[3];
  const float* w1_0 = (const float*)d_in[4];
  const float* b1_0 = (const float*)d_in[5];
  const float* w2_0 = (const float*)d_in[6];
  const float* b2_0 = (const float*)d_in[7];
  const float* w1_1 = (const float*)d_in[8];
  const float* b1_1 = (const float*)d_in[9];
  const float* w2_1 = (const float*)d_in[10];
  const float* b2_1 = (const float*)d_in[11];
  const float* w1_2 = (const float*)d_in[12];
  const float* b1_2 = (const float*)d_in[13];
  const float* w2_2 = (const float*)d_in[14];
  const float* b2_2 = (const float*)d_in[15];
  const float* w_out = (const float*)d_in[16];
  const float* b_out = (const float*)d_in[17];

  const int FIN = 128;
  const int N = in_sizes[0] / FIN;
  const int E = in_sizes[1] / 2;
  const int G = out_size;

  const int* src = eidx;      // edge_index[0]
  const int* dst = eidx + E;  // edge_index[1]

  // Workspace layout (floats): agg[N*128] | hA[N*64] | hB[N*64] | pooled[G*64]
  float* agg    = (float*)d_ws;
  float* hA     = agg + (size_t)N * FIN;
  float* hB     = hA + (size_t)N * HDIM;
  float* pooled = hB + (size_t)N * HDIM;

  const int tiles       = (N + 15) / 16;
  const int mlp_blocks  = (tiles + 7) / 8;
  const int scat_blocks = 4096;
  const int pool_blocks = 2048;

  float* out = (float*)d_out;

  // ---- layer 0 (K = 128) ----
  hipMemsetAsync(agg, 0, (size_t)N * FIN * sizeof(float), stream);
  gin_scatter_kernel<32><<<scat_blocks, 256, 0, stream>>>(x, src, dst, agg, E);
  gin_mlp_kernel<<<mlp_blocks, 256, 0, stream>>>(agg, x, w1_0, b1_0, w2_0, b2_0,
                                                 hA, N, FIN);
  // ---- layer 1 (K = 64) ----
  hipMemsetAsync(agg, 0, (size_t)N * HDIM * sizeof(float), stream);
  gin_scatter_kernel<16><<<scat_blocks, 256, 0, stream>>>(hA, src, dst, agg, E);
  gin_mlp_kernel<<<mlp_blocks, 256, 0, stream>>>(agg, hA, w1_1, b1_1, w2_1, b2_1,
                                                 hB, N, HDIM);
  // ---- layer 2 (K = 64) ----
  hipMemsetAsync(agg, 0, (size_t)N * HDIM * sizeof(float), stream);
  gin_scatter_kernel<16><<<scat_blocks, 256, 0, stream>>>(hB, src, dst, agg, E);
  gin_mlp_kernel<<<mlp_blocks, 256, 0, stream>>>(agg, hB, w1_2, b1_2, w2_2, b2_2,
                                                 hA, N, HDIM);

  // ---- pooling + readout ----
  hipMemsetAsync(pooled, 0, (size_t)G * HDIM * sizeof(float), stream);
  gin_pool_kernel<<<pool_blocks, 256, 0, stream>>>(hA, batch, pooled, N);
  gin_readout_kernel<<<(G + 255) / 256, 256, 0, stream>>>(pooled, w_out, b_out,
                                                          out, G);
}